// GraphFeatureExtractor_18537078849894
// MI455X (gfx1250) — compile-verified
//
#include <hip/hip_runtime.h>

// ---------------------------------------------------------------------------
// GraphFeatureExtractor for MI455X (gfx1250, wave32)
//   B=16, N=2048, D=3, H=128, k=16
//   Phase 1: kNN via LDS-staged point set + register insertion-sort (VALU)
//   Phase 2: 3x GraphConv = gather-sum + dual-GEMM using V_WMMA_F32_16X16X4_F32
//   Layer-1 K=3 is zero-padded to K=4 so every GEMM path is branch-free.
// ---------------------------------------------------------------------------

typedef __attribute__((ext_vector_type(2))) float v2f;
typedef __attribute__((ext_vector_type(8))) float v8f;

static constexpr int B = 16;
static constexpr int N = 2048;
static constexpr int K = 16;   // neighbors
static constexpr int H = 128;  // hidden dim

// ---------------------------------------------------------------------------
// kNN: one block of 256 threads handles 256 query rows of one batch.
// All 2048 points of the batch staged in LDS (padded to 4 floats -> 32KB).
// Each thread keeps a sorted top-16 (dist, idx) in registers.
// ---------------------------------------------------------------------------
__global__ void __launch_bounds__(256) knn_kernel(const float* __restrict__ pts,
                                                  int* __restrict__ nbr) {
  __shared__ float lp[N * 4];
  const int b  = blockIdx.x >> 3;          // 8 blocks per batch
  const int rb = (blockIdx.x & 7) << 8;    // row base within batch
  const float* p = pts + (size_t)b * N * 3;

  for (int t = threadIdx.x; t < N; t += 256) {
    lp[t * 4 + 0] = p[t * 3 + 0];
    lp[t * 4 + 1] = p[t * 3 + 1];
    lp[t * 4 + 2] = p[t * 3 + 2];
  }
  __syncthreads();

  const int i = rb + threadIdx.x;
  const float px = lp[i * 4 + 0], py = lp[i * 4 + 1], pz = lp[i * 4 + 2];

  float bd[K];
  int   bi[K];
#pragma unroll
  for (int s = 0; s < K; ++s) { bd[s] = 3.4e38f; bi[s] = 0; }

  for (int j = 0; j < N; ++j) {
    const float dx = lp[j * 4 + 0] - px;
    const float dy = lp[j * 4 + 1] - py;
    const float dz = lp[j * 4 + 2] - pz;
    const float d2 = dx * dx + dy * dy + dz * dz;
    if (d2 < bd[K - 1]) {          // rare after warmup -> cheap common path
      float v = d2; int vi = j;    // bubble-insert (branchless swap chain)
#pragma unroll
      for (int s = 0; s < K; ++s) {
        const bool lt = v < bd[s];
        const float nd = lt ? v  : bd[s];
        const int   ni = lt ? vi : bi[s];
        const float ov = lt ? bd[s] : v;
        const int   oi = lt ? bi[s] : vi;
        bd[s] = nd; bi[s] = ni; v = ov; vi = oi;
      }
    }
  }

  int* o = nbr + ((size_t)b * N + i) * K;
#pragma unroll
  for (int s = 0; s < K; ++s) o[s] = bi[s];
}

// ---------------------------------------------------------------------------
// Prep: zero-pad [rows,3] points to [rows,4] (K=4 WMMA friendly).
// ---------------------------------------------------------------------------
__global__ void __launch_bounds__(256) pad_points_kernel(const float* __restrict__ src,
                                                         float* __restrict__ dst) {
  const int row = blockIdx.x * 256 + threadIdx.x;   // rows = B*N
  float4 v;
  v.x = src[row * 3 + 0];
  v.y = src[row * 3 + 1];
  v.z = src[row * 3 + 2];
  v.w = 0.f;
  *(float4*)(dst + (size_t)row * 4) = v;
}

// Pad W1r [128,3] and W1s [128,3] into dst [2*128, 4].
__global__ void __launch_bounds__(256) pad_weights_kernel(const float* __restrict__ Wr,
                                                          const float* __restrict__ Ws,
                                                          float* __restrict__ dst) {
  const int t = threadIdx.x;             // 0..255
  const float* w = (t < 128) ? Wr : Ws;
  const int n = t & 127;
  float4 v;
  v.x = w[n * 3 + 0];
  v.y = w[n * 3 + 1];
  v.z = w[n * 3 + 2];
  v.w = 0.f;
  *(float4*)(dst + (size_t)t * 4) = v;
}

// ---------------------------------------------------------------------------
// Layer-1 aggregation: agg4[row] = sum_{kk<16} pts[b, nbr[row][kk]]  (padded 4th = 0)
// One thread per row; int4 index loads.
// ---------------------------------------------------------------------------
__global__ void __launch_bounds__(256) aggregate3_kernel(const float* __restrict__ pts,
                                                         const int* __restrict__ nbr,
                                                         float* __restrict__ agg4) {
  const int row = blockIdx.x * 256 + threadIdx.x;   // rows = B*N
  const int b   = row >> 11;                        // N = 2048
  const int* nb = nbr + (size_t)row * K;
  const float* xb = pts + (size_t)b * N * 3;

  int idxv[K];
#pragma unroll
  for (int q = 0; q < 4; ++q) {
    const int4 iv = *(const int4*)(nb + q * 4);
    idxv[q * 4 + 0] = iv.x; idxv[q * 4 + 1] = iv.y;
    idxv[q * 4 + 2] = iv.z; idxv[q * 4 + 3] = iv.w;
  }
  float sx = 0.f, sy = 0.f, sz = 0.f;
#pragma unroll
  for (int kk = 0; kk < K; ++kk) {
    const float* pj = xb + (size_t)idxv[kk] * 3;
    sx += pj[0]; sy += pj[1]; sz += pj[2];
  }
  float4 v; v.x = sx; v.y = sy; v.z = sz; v.w = 0.f;
  *(float4*)(agg4 + (size_t)row * 4) = v;
}

// ---------------------------------------------------------------------------
// H=128 aggregation: one wave per row, lanes = 32 channel-quads.
// float4 gathers -> coalesced 512B per neighbor per wave.
// ---------------------------------------------------------------------------
__global__ void __launch_bounds__(256) aggregate_h_kernel(const float* __restrict__ x,
                                                          const int* __restrict__ nbr,
                                                          float* __restrict__ agg) {
  const int idx = blockIdx.x * 256 + threadIdx.x;   // rows * 32
  const int row = idx >> 5;
  const int c4  = (idx & 31) << 2;                  // channel base (0,4,...,124)
  const int b   = row >> 11;
  const int* nb = nbr + (size_t)row * K;
  const float* xb = x + (size_t)b * N * H;

  int idxv[K];
#pragma unroll
  for (int q = 0; q < 4; ++q) {
    const int4 iv = *(const int4*)(nb + q * 4);
    idxv[q * 4 + 0] = iv.x; idxv[q * 4 + 1] = iv.y;
    idxv[q * 4 + 2] = iv.z; idxv[q * 4 + 3] = iv.w;
  }
  float sx = 0.f, sy = 0.f, sz = 0.f, sw = 0.f;
#pragma unroll
  for (int kk = 0; kk < K; ++kk) {
    const float4 v = *(const float4*)(xb + (size_t)idxv[kk] * H + c4);
    sx += v.x; sy += v.y; sz += v.z; sw += v.w;
  }
  float4 s; s.x = sx; s.y = sy; s.z = sz; s.w = sw;
  *(float4*)(agg + (size_t)row * H + c4) = s;
}

// ---------------------------------------------------------------------------
// Dual GEMM: out[M=B*N, 128] = agg @ Wr^T + x @ Ws^T + bias  (+ReLU)
// fp32 exact via V_WMMA_F32_16X16X4_F32; DIN is 4 or 128 (always K%4==0).
// One wave -> 16 rows x 128 cols (8 v8f accumulators).
// Per K-step: batch all 18 v2f (b64) loads, then 16 back-to-back WMMAs.
// A layout (16x4 f32): lanes 0-15 M=lane, K={k0,k0+1}; lanes 16-31 K={k0+2,k0+3}
// B layout (4x16 f32): mirrored; B[k][n] = W[n*DIN + k]
// C/D (16x16 f32): VGPR r = rows m0+r (lanes 0-15) and m0+8+r (lanes 16-31)
// ---------------------------------------------------------------------------
template <int DIN, bool RELU>
__global__ void __launch_bounds__(256) gemm_dual_kernel(
    const float* __restrict__ agg, const float* __restrict__ x,
    const float* __restrict__ Wr,  const float* __restrict__ Ws,
    const float* __restrict__ bias, float* __restrict__ out) {
  static_assert((DIN & 3) == 0, "DIN must be a multiple of 4");
  const int lane  = threadIdx.x & 31;
  const int wave  = threadIdx.x >> 5;
  const int m0    = (blockIdx.x * 8 + wave) * 16;
  const int mrow  = m0 + (lane & 15);
  const int khalf = ((lane >> 4) << 1);   // 0 or 2
  const int ncol  = lane & 15;

  const float* aRp = agg + (size_t)mrow * DIN + khalf;
  const float* aXp = x   + (size_t)mrow * DIN + khalf;

  v8f acc[8];
  const v8f zero = {0.f, 0.f, 0.f, 0.f, 0.f, 0.f, 0.f, 0.f};
#pragma unroll
  for (int nt = 0; nt < 8; ++nt) acc[nt] = zero;

#pragma unroll 2
  for (int k0 = 0; k0 < DIN; k0 += 4) {
    // ---- batch all loads for this K-step (8B each, 8B-aligned) ----
    const v2f aR = *(const v2f*)(aRp + k0);
    const v2f aX = *(const v2f*)(aXp + k0);
    v2f bR[8], bS[8];
#pragma unroll
    for (int nt = 0; nt < 8; ++nt) {
      const size_t wo = (size_t)(nt * 16 + ncol) * DIN + khalf + k0;
      bR[nt] = *(const v2f*)(Wr + wo);
      bS[nt] = *(const v2f*)(Ws + wo);
    }
    // ---- dense WMMA clause ----
#pragma unroll
    for (int nt = 0; nt < 8; ++nt)
      acc[nt] = __builtin_amdgcn_wmma_f32_16x16x4_f32(
          false, aR, false, bR[nt], (short)0, acc[nt], false, false);
#pragma unroll
    for (int nt = 0; nt < 8; ++nt)
      acc[nt] = __builtin_amdgcn_wmma_f32_16x16x4_f32(
          false, aX, false, bS[nt], (short)0, acc[nt], false, false);
  }

  // Epilogue: bias (+ReLU), scatter per C/D layout.
  const int rbase = m0 + ((lane >> 4) << 3);  // lanes 16-31 -> rows +8
#pragma unroll
  for (int nt = 0; nt < 8; ++nt) {
    const int n = nt * 16 + ncol;
    const float bv = bias[n];
#pragma unroll
    for (int r = 0; r < 8; ++r) {
      float v = acc[nt][r] + bv;
      if (RELU) v = fmaxf(v, 0.f);
      out[(size_t)(rbase + r) * H + n] = v;
    }
  }
}

// ---------------------------------------------------------------------------
// Launch. Workspace layout (bytes):
//   [0,2M)    nbr      B*N*K int32
//   [2M,18M)  agg      B*N*H f32
//   [18M,34M) x1       B*N*H f32
//   [34M,50M) x2       B*N*H f32
//   [50M,50.5M) pts4   B*N*4 f32
//   [50.5M,51M) agg4   B*N*4 f32
//   [51M, +4K)  w14    2*128*4 f32 (padded W1r | W1s)
// ---------------------------------------------------------------------------
extern "C" void kernel_launch(void* const* d_in, const int* in_sizes, int n_in,
                              void* d_out, int out_size, void* d_ws, size_t ws_size,
                              hipStream_t stream) {
  const float* pts = (const float*)d_in[0];
  const float* W1r = (const float*)d_in[1];
  const float* b1  = (const float*)d_in[2];
  const float* W1s = (const float*)d_in[3];
  const float* W2r = (const float*)d_in[4];
  const float* b2  = (const float*)d_in[5];
  const float* W2s = (const float*)d_in[6];
  const float* W3r = (const float*)d_in[7];
  const float* b3  = (const float*)d_in[8];
  const float* W3s = (const float*)d_in[9];
  (void)in_sizes; (void)n_in; (void)out_size; (void)ws_size;

  char*  ws   = (char*)d_ws;
  int*   nbr  = (int*)ws;
  float* agg  = (float*)(ws + ((size_t)2  << 20));
  float* x1   = (float*)(ws + ((size_t)18 << 20));
  float* x2   = (float*)(ws + ((size_t)34 << 20));
  float* pts4 = (float*)(ws + ((size_t)50 << 20));
  float* agg4 = (float*)(ws + ((size_t)50 << 20) + ((size_t)512 << 10));
  float* w14  = (float*)(ws + ((size_t)51 << 20));
  float* out  = (float*)d_out;

  const int rows = B * N;                   // 32768
  const int row_blocks  = rows / 256;       // 128
  const int gemm_blocks = rows / (8 * 16);  // 256 (8 waves/block, 16 rows/wave)

  // 1) kNN graph + layer-1 padding prep (independent, back-to-back issue)
  knn_kernel<<<B * 8, 256, 0, stream>>>(pts, nbr);
  pad_points_kernel<<<row_blocks, 256, 0, stream>>>(pts, pts4);
  pad_weights_kernel<<<1, 256, 0, stream>>>(W1r, W1s, w14);

  // 2) Layer 1: din=3 (padded to 4) -> 128, ReLU
  aggregate3_kernel<<<row_blocks, 256, 0, stream>>>(pts, nbr, agg4);
  gemm_dual_kernel<4, true><<<gemm_blocks, 256, 0, stream>>>(
      agg4, pts4, w14, w14 + 128 * 4, b1, x1);

  // 3) Layer 2: 128 -> 128, ReLU
  aggregate_h_kernel<<<rows * 32 / 256, 256, 0, stream>>>(x1, nbr, agg);
  gemm_dual_kernel<128, true><<<gemm_blocks, 256, 0, stream>>>(agg, x1, W2r, W2s, b2, x2);

  // 4) Layer 3: 128 -> 128, no ReLU, straight to d_out
  aggregate_h_kernel<<<rows * 32 / 256, 256, 0, stream>>>(x2, nbr, agg);
  gemm_dual_kernel<128, false><<<gemm_blocks, 256, 0, stream>>>(agg, x2, W3r, W3s, b3, out);
}